// GNNWithFragments_29918742184480
// MI455X (gfx1250) — compile-verified
//
#include <hip/hip_runtime.h>
#include <hip/hip_bf16.h>
#include <math.h>

// Problem constants (match reference).
#define NN 50000
#define EE 800000
#define FF 64
#define HH 64
#define NFRAG 1024

typedef float v2f __attribute__((ext_vector_type(2)));
typedef float v8f __attribute__((ext_vector_type(8)));

// ---------------------------------------------------------------- utilities
__global__ void fillz_kernel(float* __restrict__ p, int n) {
    int i = blockIdx.x * blockDim.x + threadIdx.x;
    if (i < n) p[i] = 0.0f;
}

// ------------------------------------------------------- edge scatter-add
// agg[row[e], :] += src[col[e], :]   (16 threads per edge, float4 per thread)
__global__ void scatter_kernel(const float* __restrict__ src,
                               const long long* __restrict__ ei,
                               float* __restrict__ agg, int nedges) {
    int tid = blockIdx.x * blockDim.x + threadIdx.x;
    int e = tid >> 4;
    if (e >= nedges) return;
    int fq = (tid & 15) << 2;                 // feature quad: 0,4,...,60
    long long r = ei[e];                      // destination row
    long long c = ei[nedges + e];             // source row
    const float4 v = *(const float4*)(src + c * 64 + fq);
    float* dst = agg + r * 64 + fq;
    atomicAdd(dst + 0, v.x);
    atomicAdd(dst + 1, v.y);
    atomicAdd(dst + 2, v.z);
    atomicAdd(dst + 3, v.w);
}

// --------------------------------------------- WMMA fp32 GEMM + bias + ReLU
// out[n,64] = relu(A[n,64] @ W^T + b), W is [64 out][64 in], fp32 throughout.
// One wave = one 16x16 output tile; 8 waves/block cover 32 rows x 64 cols.
// K=64 accumulated as 16 chained V_WMMA_F32_16X16X4_F32.
__global__ void gemm_relu_wmma(const float* __restrict__ A,
                               const float* __restrict__ W,
                               const float* __restrict__ bias,
                               float* __restrict__ out, int n) {
    __shared__ float Bs[64 * 65];             // Bs[k*65 + j] = W[j][k]; pad kills bank conflicts
    const int tid = threadIdx.x;

    // Cooperative transpose load of the 64x64 weight tile (coalesced global reads).
    for (int idx = tid; idx < 64 * 64; idx += 256) {
        int j = idx >> 6;                     // output index
        int k = idx & 63;                     // input index
        Bs[k * 65 + j] = W[idx];
    }
    __syncthreads();

    const int wid  = tid >> 5;
    const int lane = tid & 31;
    const int m0 = blockIdx.x * 32 + (wid >> 2) * 16;   // row-tile base
    if (m0 >= n) return;                                 // (after barrier; safe)
    const int j0   = (wid & 3) << 4;                     // col-tile base
    const int half = lane >> 4;                          // 0: lanes 0-15, 1: lanes 16-31
    const int mr   = lane & 15;

    v8f c = {};
    // A fragment base: lane holds row (m0+mr), K pair {2*half, 2*half+1} per step.
    const float* arow = A + (size_t)(m0 + mr) * 64 + 2 * half;

#pragma unroll
    for (int kk = 0; kk < 16; ++kk) {
        // A 16x4 f32 fragment (ISA layout): v0=K(4kk+2h), v1=K(4kk+2h+1)
        v2f a = *(const v2f*)(arow + kk * 4);
        int krow = kk * 4 + 2 * half;
        v2f b;
        b.x = Bs[(krow + 0) * 65 + j0 + mr];             // B row K, cols j0..j0+15
        b.y = Bs[(krow + 1) * 65 + j0 + mr];
        c = __builtin_amdgcn_wmma_f32_16x16x4_f32(
                /*neg_a=*/false, a, /*neg_b=*/false, b,
                /*c_mod=*/(short)0, c, /*reuse_a=*/false, /*reuse_b=*/false);
    }

    // D layout: VGPR r -> M = m0 + r + 8*half, N = j0 + mr. Bias + ReLU + store.
    const float bv = bias[j0 + mr];
    const int col = j0 + mr;
#pragma unroll
    for (int r = 0; r < 8; ++r) {
        int row = m0 + r + 8 * half;
        float v = c[r] + bv;
        out[(size_t)row * 64 + col] = v > 0.0f ? v : 0.0f;
    }
}

// ------------------------------------------------------------ column sums
// gsum[f] += sum over strided rows of xin[r, f]   (mean applied in head)
__global__ void colsum_kernel(const float* __restrict__ xin,
                              float* __restrict__ gsum, int n) {
    int f = threadIdx.x & 63;
    int rpb = blockDim.x >> 6;                 // rows per block per iter (4)
    int r0 = blockIdx.x * rpb + (threadIdx.x >> 6);
    int stride = gridDim.x * rpb;
    float s = 0.0f;
    for (int r = r0; r < n; r += stride) s += xin[(size_t)r * 64 + f];
    atomicAdd(&gsum[f], s);
}

// ----------------------------------------------------------------- MLP head
// h = relu(Wf1 @ concat(gsum/N, frag) + bf1); out = sigmoid(Wf2 @ h + bf2)
__global__ void head_kernel(const float* __restrict__ gsum,
                            const float* __restrict__ frag,
                            const float* __restrict__ Wf1,
                            const float* __restrict__ bf1,
                            const float* __restrict__ Wf2,
                            const float* __restrict__ bf2,
                            float* __restrict__ out, float invN) {
    __shared__ float h[32];
    const int tid = threadIdx.x;               // 256 threads
    const int oi = tid >> 3;                   // output unit 0..31
    const int part = tid & 7;                  // 8-way split of the 1088-dot
    float s = 0.0f;
    for (int idx = part; idx < HH + NFRAG; idx += 8) {
        float v = (idx < HH) ? gsum[idx] * invN : frag[idx - HH];
        s += Wf1[oi * (HH + NFRAG) + idx] * v;
    }
    s += __shfl_down(s, 4, 8);
    s += __shfl_down(s, 2, 8);
    s += __shfl_down(s, 1, 8);
    if (part == 0) {
        float v = s + bf1[oi];
        h[oi] = v > 0.0f ? v : 0.0f;
    }
    __syncthreads();
    if (tid == 0) {
        float o = bf2[0];
#pragma unroll
        for (int j = 0; j < 32; ++j) o += Wf2[j] * h[j];
        out[0] = 1.0f / (1.0f + expf(-o));
    }
}

// ------------------------------------------------------------------ launch
extern "C" void kernel_launch(void* const* d_in, const int* in_sizes, int n_in,
                              void* d_out, int out_size, void* d_ws, size_t ws_size,
                              hipStream_t stream) {
    (void)in_sizes; (void)n_in; (void)out_size; (void)ws_size;

    const float*     x    = (const float*)d_in[0];
    const long long* ei   = (const long long*)d_in[1];   // int64 [2, E]
    const float*     frag = (const float*)d_in[2];
    const float*     W1   = (const float*)d_in[3];
    const float*     b1   = (const float*)d_in[4];
    const float*     W2   = (const float*)d_in[5];
    const float*     b2   = (const float*)d_in[6];
    const float*     W3   = (const float*)d_in[7];
    const float*     b3   = (const float*)d_in[8];
    const float*     Wf1  = (const float*)d_in[9];
    const float*     bf1  = (const float*)d_in[10];
    const float*     Wf2  = (const float*)d_in[11];
    const float*     bf2  = (const float*)d_in[12];
    float* out = (float*)d_out;

    // Workspace: agg[N*64] | xbuf[N*64] | gsum[64]
    float* agg  = (float*)d_ws;
    float* xbuf = agg + (size_t)NN * 64;
    float* gsum = xbuf + (size_t)NN * 64;

    const int nfeat = NN * 64;
    const dim3 fillGrid((nfeat + 255) / 256);
    const dim3 scatGrid((EE * 16 + 255) / 256);
    const dim3 gemmGrid((NN + 31) / 32);

    // Layer 1: agg = scatter(x); xbuf = relu(agg @ W1^T + b1)
    fillz_kernel<<<fillGrid, 256, 0, stream>>>(agg, nfeat);
    scatter_kernel<<<scatGrid, 256, 0, stream>>>(x, ei, agg, EE);
    gemm_relu_wmma<<<gemmGrid, 256, 0, stream>>>(agg, W1, b1, xbuf, NN);

    // Layer 2
    fillz_kernel<<<fillGrid, 256, 0, stream>>>(agg, nfeat);
    scatter_kernel<<<scatGrid, 256, 0, stream>>>(xbuf, ei, agg, EE);
    gemm_relu_wmma<<<gemmGrid, 256, 0, stream>>>(agg, W2, b2, xbuf, NN);

    // Layer 3
    fillz_kernel<<<fillGrid, 256, 0, stream>>>(agg, nfeat);
    scatter_kernel<<<scatGrid, 256, 0, stream>>>(xbuf, ei, agg, EE);
    gemm_relu_wmma<<<gemmGrid, 256, 0, stream>>>(agg, W3, b3, xbuf, NN);

    // Mean + head
    fillz_kernel<<<1, 64, 0, stream>>>(gsum, 64);
    colsum_kernel<<<512, 256, 0, stream>>>(xbuf, gsum, NN);
    head_kernel<<<1, 256, 0, stream>>>(gsum, frag, Wf1, bf1, Wf2, bf2, out,
                                       1.0f / (float)NN);
}